// FrechtMean_86612310492073
// MI455X (gfx1250) — compile-verified
//
#include <hip/hip_runtime.h>
#include <hip/hip_bf16.h>
#include <stdint.h>

typedef __attribute__((ext_vector_type(2))) float v2f;
typedef __attribute__((ext_vector_type(8))) float v8f;

#define P_PTS 128   // points per batch
#define D_DIM 64    // embedding dim

// One workgroup (128 threads = 4 waves) per batch.
// K = -1 hardcoded as in the reference.
__global__ __launch_bounds__(128) void frechet_mean_kernel(
    const float* __restrict__ X, const float* __restrict__ w,
    const int* __restrict__ max_iter, float* __restrict__ out)
{
    __shared__ float Xs[P_PTS][D_DIM];   // 32 KB, row-major copy of X[b]
    __shared__ float alphas_sh[P_PTS];
    __shared__ float mu_sh[D_DIM];
    __shared__ float bvec[D_DIM];
    __shared__ float redA[4], redC[4];
    __shared__ float muss0_sh;

    const int tid  = threadIdx.x;        // == point index p
    const int lane = tid & 31;
    const int wid  = tid >> 5;
    const bool hi  = lane >= 16;
    const int m    = lane & 15;          // M row within 16x16 tile
    const int d0   = wid * 16;           // this wave's dim tile

    const float* Xg = X + (size_t)blockIdx.x * (P_PTS * D_DIM);

    // ---- async global -> LDS copy of the 32 KB X tile (ASYNCcnt path) ----
    unsigned ldsBase = (unsigned)(uintptr_t)(&Xs[0][0]);
    #pragma unroll
    for (int i = 0; i < 16; ++i) {
        unsigned goff  = (unsigned)(tid + i * 128) * 16u;  // float4 index -> bytes
        unsigned laddr = ldsBase + goff;
        asm volatile("global_load_async_to_lds_b128 %0, %1, %2"
                     :: "v"(laddr), "v"(goff), "s"(Xg) : "memory");
    }
    asm volatile("s_wait_asynccnt 0" ::: "memory");
    __syncthreads();

    // ---- x_ss[p] (bank-rotated, conflict-free) ----
    float x2 = 0.f;
    #pragma unroll
    for (int j = 0; j < 64; ++j) {
        int d = (lane + j) & 63;
        float v = Xs[tid][d];
        x2 = fmaf(v, v, x2);
    }
    if (tid == 0) muss0_sh = x2;                 // |X[0]|^2 = initial mu_ss
    if (tid < D_DIM) mu_sh[tid] = Xs[0][tid];    // mu0 = X[b,0,:]
    float wt = w[(size_t)blockIdx.x * P_PTS + tid];
    __syncthreads();

    float mu_ss = muss0_sh;
    int iters = max_iter[0];
    if (iters < 0) iters = 0;

    for (int it = 0; it < iters; ++it) {
        // (1) dot[p] = X[p] . mu   (rotated LDS walk, conflict-free)
        float dot = 0.f;
        #pragma unroll
        for (int j = 0; j < 64; ++j) {
            int d = (lane + j) & 63;
            dot = fmaf(Xs[tid][d], mu_sh[d], dot);
        }

        // (2) alpha[p]  (K = -1)
        float xmu   = x2 - 2.0f * dot + mu_ss;          // |X_p - mu|^2
        float denom = (1.0f - x2) * (1.0f - mu_ss);
        float y     = fmaxf(xmu / denom, 1e-6f);
        float xx    = fmaxf(1.0f + 2.0f * y, 1.0f + 1e-6f);
        float z     = sqrtf(fmaf(xx, xx, -1.0f));
        float lp    = 2.0f * logf(xx + z) / sqrtf(fmaf(y, y, y));
        float alpha = lp / (1.0f - x2) * wt;
        alphas_sh[tid] = alpha;

        // (3) scalar reductions a = sum(alpha), c = sum(alpha*x_ss)
        float av = alpha, cv = alpha * x2;
        #pragma unroll
        for (int off = 16; off > 0; off >>= 1) {
            av += __shfl_down(av, off, 32);
            cv += __shfl_down(cv, off, 32);
        }
        if (lane == 0) { redA[wid] = av; redC[wid] = cv; }
        __syncthreads();

        // (4) b[d0..d0+15] = sum_p alpha[p] * X[p][d]  via V_WMMA_F32_16X16X4_F32
        //     A(16x4) = X^T sub-tile  : A[m][k] = Xs[p0+k][d0+m]
        //       lanes 0-15: VGPR0=K0, VGPR1=K1; lanes 16-31: VGPR0=K2, VGPR1=K3
        //     B(4x16)  = alpha broadcast over N: B[k][n] = alphas[p0+k]
        //       (same K-per-lane-half pattern as A)
        v8f acc = {};
        #pragma unroll
        for (int p0 = 0; p0 < P_PTS; p0 += 4) {
            int k0 = p0 + (hi ? 2 : 0);
            v2f Av, Bv;
            Av.x = Xs[k0][d0 + m];
            Av.y = Xs[k0 + 1][d0 + m];
            Bv.x = alphas_sh[k0];
            Bv.y = alphas_sh[k0 + 1];
            acc = __builtin_amdgcn_wmma_f32_16x16x4_f32(
                      false, Av, false, Bv, (short)0, acc, false, false);
        }
        // Every output column is identical; C/D layout: VGPR r -> M=r (lanes 0-15),
        // M=r+8 (lanes 16-31). Lane 0 / lane 16 scatter the 16 dims.
        if (lane == 0 || lane == 16) {
            int dbase = d0 + (hi ? 8 : 0);
            #pragma unroll
            for (int r = 0; r < 8; ++r) bvec[dbase + r] = acc[r];
        }
        __syncthreads();

        // (5) eta and mu update (all scalars uniform across the block)
        float a = redA[0] + redA[1] + redA[2] + redA[3];
        float c = redC[0] + redC[1] + redC[2] + redC[3];
        float bss = 0.f;
        #pragma unroll
        for (int d = 0; d < D_DIM; ++d) {   // broadcast LDS reads
            float bd = bvec[d];
            bss = fmaf(bd, bd, bss);
        }
        float akc = a + c;                                   // a - K*c, K=-1
        float eta = (akc - sqrtf(fmaf(akc, akc, -4.0f * bss))) / (2.0f * bss);
        if (tid < D_DIM) mu_sh[tid] = eta * bvec[tid];
        mu_ss = eta * eta * bss;                             // |mu_new|^2 analytically
        __syncthreads();
    }

    if (tid < D_DIM) out[(size_t)blockIdx.x * D_DIM + tid] = mu_sh[tid];
}

extern "C" void kernel_launch(void* const* d_in, const int* in_sizes, int n_in,
                              void* d_out, int out_size, void* d_ws, size_t ws_size,
                              hipStream_t stream) {
    const float* X  = (const float*)d_in[0];
    const float* w  = (const float*)d_in[1];
    const int*   mi = (const int*)d_in[2];
    float*       out = (float*)d_out;
    (void)n_in; (void)out_size; (void)d_ws; (void)ws_size;

    int B = in_sizes[0] / (P_PTS * D_DIM);   // 1024
    frechet_mean_kernel<<<dim3(B), dim3(P_PTS), 0, stream>>>(X, w, mi, out);
}